// TransR_84911503442474
// MI455X (gfx1250) — compile-verified
//
#include <hip/hip_runtime.h>
#include <hip/hip_bf16.h>

#define ENT_TOTAL 100000
#define REL_TOTAL 1000
#define ENT_SIZE  200
#define REL_SIZE  200
#define BATCH     16384
#define MARGIN    4.0f

#define NPAD   208   // projection buffer pitch (13*16 columns)
#define NCT    13    // column tiles of 16
#define APITCH 204   // A tile row pitch (bank-conflict friendly)
#define MPITCH 200   // M (transfer matrix) natural pitch for TDM staging

typedef __attribute__((ext_vector_type(2))) float v2f;
typedef __attribute__((ext_vector_type(8))) float v8f;
typedef __attribute__((ext_vector_type(4))) unsigned int u32x4;
typedef __attribute__((ext_vector_type(8))) int i32x8;
typedef __attribute__((ext_vector_type(4))) int i32x4;

// ---------------- bucketing kernels ----------------

__global__ __launch_bounds__(256) void zero_counts(int* __restrict__ counts) {
    int i = blockIdx.x * 256 + threadIdx.x;
    if (i < 1024) counts[i] = 0;
}

__global__ __launch_bounds__(256) void count_rel(const int* __restrict__ pos_r,
                                                 int* __restrict__ counts) {
    int i = blockIdx.x * 256 + threadIdx.x;
    if (i < BATCH) atomicAdd(&counts[pos_r[i]], 1);
}

__global__ __launch_bounds__(1024) void scan_rel(const int* __restrict__ counts,
                                                 int* __restrict__ offsets,
                                                 int* __restrict__ cursor) {
    __shared__ int s[1024];
    int t = threadIdx.x;
    int own = (t < REL_TOTAL) ? counts[t] : 0;
    s[t] = own;
    __syncthreads();
    for (int off = 1; off < 1024; off <<= 1) {
        int v = (t >= off) ? s[t - off] : 0;
        __syncthreads();
        s[t] += v;
        __syncthreads();
    }
    int excl = s[t] - own;
    if (t < REL_TOTAL) { offsets[t] = excl; cursor[t] = excl; }
}

__global__ __launch_bounds__(256) void scatter_rel(const int* __restrict__ pos_r,
                                                   int* __restrict__ cursor,
                                                   int* __restrict__ bucket) {
    int i = blockIdx.x * 256 + threadIdx.x;
    if (i < BATCH) {
        int r = pos_r[i];
        int p = atomicAdd(&cursor[r], 1);
        bucket[p] = i;
    }
}

// ---------------- main per-relation kernel ----------------
// One workgroup per relation r. All bucketed batch elements of this block
// share transfer[r] (staged once in LDS via the Tensor Data Mover) and
// rel_emb[r].

__global__ __launch_bounds__(256) void transr_rel_kernel(
    const int* __restrict__ pos_h, const int* __restrict__ pos_t,
    const int* __restrict__ neg_h, const int* __restrict__ neg_t,
    const float* __restrict__ ent_emb, const float* __restrict__ rel_emb,
    const float* __restrict__ transfer,
    const int* __restrict__ offsets, const int* __restrict__ counts,
    const int* __restrict__ bucket, float* __restrict__ scores)
{
    __shared__ float Mlds[ENT_SIZE * MPITCH + 16]; // 160 KB M + slack for tail over-read
    __shared__ float Als[16 * APITCH];             // A tile: 16 entity vectors
    __shared__ float Pls[16 * NPAD];               // projection results
    __shared__ float Rn[REL_SIZE];                 // normalized relation vec
    __shared__ float red[256];
    __shared__ float red2[256];
    __shared__ float invn[16];

    const int r    = blockIdx.x;
    const int tid  = threadIdx.x;
    const int lane = tid & 31;
    const int wave = tid >> 5;
    const int cnt  = counts[r];
    const int seg  = offsets[r];

    // ---- TDM: DMA transfer[r] (200x200 f32, row-major) into Mlds ----
    // D# per CDNA5 ISA: 2-D tensor, tile == tensor,
    // groups 2/3 zero (tile_dim2 = 0 => unused). Issued by wave 0 only.
    const float* Mg = transfer + (size_t)r * (ENT_SIZE * REL_SIZE);
    if (wave == 0) {
        unsigned long long ga = (unsigned long long)(uintptr_t)Mg;
        unsigned lds_off = (unsigned)(uintptr_t)(void*)&Mlds[0]; // flat LDS addr[31:0] = LDS offset

        u32x4 g0;
        g0.x = 1u;                                   // count=1, user descriptor
        g0.y = lds_off;                              // lds_addr (bytes)
        g0.z = (unsigned)(ga & 0xFFFFFFFFu);         // global_addr[31:0]
        g0.w = (unsigned)((ga >> 32) & 0x01FFFFFFu)  // global_addr[56:32]
             | (2u << 30);                           // type = 2 ("image")

        i32x8 g1;
        g1[0] = (int)(2u << 16);          // wg_mask=0, data_size=2 (4 bytes), no pad/iter
        g1[1] = (int)(200u << 16);        // atomic_barrier_addr=0 | tensor_dim0[15:0]=200
        g1[2] = (int)(200u << 16);        // tensor_dim0[31:16]=0  | tensor_dim1[15:0]=200
        g1[3] = (int)(200u << 16);        // tensor_dim1[31:16]=0  | tile_dim0=200
        g1[4] = (int)(200u);              // tile_dim1=200 | tile_dim2=0
        g1[5] = (int)(200u);              // tensor_dim0_stride[31:0]=200
        g1[6] = 0;                        // dim0_stride[47:32]=0 | dim1_stride[15:0]=0
        g1[7] = 0;                        // dim1_stride[47:16]=0

        i32x4 gz4 = {0, 0, 0, 0};
        i32x8 gz8 = {0, 0, 0, 0, 0, 0, 0, 0};
        __builtin_amdgcn_tensor_load_to_lds(g0, g1, gz4, gz4, gz8, 0);
    }

    // ---- normalized relation embedding (overlaps with the TDM transfer) ----
    {
        float ss = 0.0f;
        for (int j = tid; j < REL_SIZE; j += 256) {
            float v = rel_emb[r * REL_SIZE + j];
            ss += v * v;
        }
        red[tid] = ss;
        __syncthreads();
        for (int s2 = 128; s2 > 0; s2 >>= 1) {
            if (tid < s2) red[tid] += red[tid + s2];
            __syncthreads();
        }
        float inv = rsqrtf(fmaxf(red[0], 1e-12f));
        for (int j = tid; j < REL_SIZE; j += 256)
            Rn[j] = rel_emb[r * REL_SIZE + j] * inv;
    }

    // issuing wave drains TENSORcnt, then barrier publishes Mlds to all waves
    if (wave == 0) __builtin_amdgcn_s_wait_tensorcnt(0);
    __syncthreads();

    if (cnt == 0) return;
    const int nvec = 4 * cnt;              // 4 vectors per element
    const int nrt  = (nvec + 15) >> 4;     // row tiles of 16

    for (int rt = 0; rt < nrt; ++rt) {
        // ---- stage A tile: rows = {pos_h,pos_t,neg_h,neg_t} x 4 elements ----
        for (int i = tid; i < 16 * ENT_SIZE; i += 256) {
            int row = i / ENT_SIZE;
            int e   = i - row * ENT_SIZE;
            int vi  = rt * 16 + row;
            float v = 0.0f;
            if (vi < nvec) {
                int el    = bucket[seg + (vi >> 2)];
                int which = vi & 3;
                int idx   = (which == 0) ? pos_h[el]
                          : (which == 1) ? pos_t[el]
                          : (which == 2) ? neg_h[el]
                          :                neg_t[el];
                v = ent_emb[(size_t)idx * ENT_SIZE + e];
            }
            Als[row * APITCH + e] = v;
        }
        __syncthreads();

        // ---- WMMA sweep: proj[16 x 200] = A[16 x 200] * M[200 x 200] ----
        // (col tiles 0..12; tile 12 columns 200..207 compute garbage from the
        //  next M row -- never read downstream, norms/scores use j < 200)
        {
            const int arow  = lane & 15;          // A row / B col within tile
            const int khalf = (lane >> 4) * 2;    // K sub-offset: 0 or 2
            for (int ct = wave; ct < NCT; ct += 8) {
                const int ncol = ct * 16 + (lane & 15);
                v8f acc = {};
                for (int k0 = 0; k0 < ENT_SIZE; k0 += 4) {
                    const int kk = k0 + khalf;
                    v2f a = *(const v2f*)&Als[arow * APITCH + kk];
                    v2f b;
                    b.x = Mlds[kk * MPITCH + ncol];
                    b.y = Mlds[(kk + 1) * MPITCH + ncol];
                    acc = __builtin_amdgcn_wmma_f32_16x16x4_f32(
                        false, a, false, b, (short)0, acc, false, false);
                }
                const int mbase = (lane >> 4) * 8; // C/D: VGPR i -> row i or i+8
                #pragma unroll
                for (int i = 0; i < 8; ++i)
                    Pls[(mbase + i) * NPAD + ncol] = acc[i];
            }
        }
        __syncthreads();

        // ---- per-row inverse L2 norms over the 200 valid columns ----
        {
            int row  = tid & 15;
            int part = tid >> 4;
            float ss = 0.0f;
            for (int j = part; j < REL_SIZE; j += 16) {
                float v = Pls[row * NPAD + j];
                ss += v * v;
            }
            red[tid] = ss;
            __syncthreads();
            for (int s2 = 128; s2 >= 16; s2 >>= 1) {
                if (tid < s2) red[tid] += red[tid + s2];
                __syncthreads();
            }
            if (tid < 16) invn[tid] = rsqrtf(fmaxf(red[tid], 1e-12f));
        }
        __syncthreads();

        // ---- scoring: 4 elements x 64 threads each ----
        {
            int el  = tid >> 6;
            int t64 = tid & 63;
            int q   = rt * 4 + el;
            float ps = 0.0f, ns = 0.0f;
            if (q < cnt) {
                const float iph = invn[4 * el + 0];
                const float ipt = invn[4 * el + 1];
                const float inh = invn[4 * el + 2];
                const float int_ = invn[4 * el + 3];
                const float* Ph = &Pls[(4 * el + 0) * NPAD];
                const float* Pt = &Pls[(4 * el + 1) * NPAD];
                const float* Nh = &Pls[(4 * el + 2) * NPAD];
                const float* Nt = &Pls[(4 * el + 3) * NPAD];
                for (int j = t64; j < REL_SIZE; j += 64) {
                    float rn = Rn[j];
                    ps += fabsf(Ph[j] * iph + rn - Pt[j] * ipt);
                    ns += fabsf(Nh[j] * inh + rn - Nt[j] * int_);
                }
            }
            red[tid]  = ps;
            red2[tid] = ns;
            __syncthreads();
            for (int s2 = 32; s2 > 0; s2 >>= 1) {
                if (t64 < s2) {
                    red[tid]  += red[tid + s2];
                    red2[tid] += red2[tid + s2];
                }
                __syncthreads();
            }
            if (t64 == 0 && q < cnt) {
                float m = fmaxf(red[tid] - red2[tid] + MARGIN, 0.0f);
                scores[bucket[seg + q]] = m;
            }
        }
        __syncthreads();
    }
}

// ---------------- deterministic mean reduction ----------------

__global__ __launch_bounds__(256) void partial_sum(const float* __restrict__ scores,
                                                   float* __restrict__ partials) {
    __shared__ float red[256];
    int i = blockIdx.x * 256 + threadIdx.x;
    red[threadIdx.x] = scores[i];
    __syncthreads();
    for (int s2 = 128; s2 > 0; s2 >>= 1) {
        if (threadIdx.x < s2) red[threadIdx.x] += red[threadIdx.x + s2];
        __syncthreads();
    }
    if (threadIdx.x == 0) partials[blockIdx.x] = red[0];
}

__global__ void final_sum(const float* __restrict__ partials, float* __restrict__ out) {
    if (threadIdx.x == 0) {
        float s = 0.0f;
        for (int i = 0; i < 64; ++i) s += partials[i];
        out[0] = s * (1.0f / (float)BATCH);
    }
}

// ---------------- launcher ----------------

extern "C" void kernel_launch(void* const* d_in, const int* in_sizes, int n_in,
                              void* d_out, int out_size, void* d_ws, size_t ws_size,
                              hipStream_t stream) {
    (void)in_sizes; (void)n_in; (void)out_size; (void)ws_size;

    const int*   pos_h = (const int*)d_in[0];
    const int*   pos_t = (const int*)d_in[1];
    const int*   pos_r = (const int*)d_in[2];
    const int*   neg_h = (const int*)d_in[3];
    const int*   neg_t = (const int*)d_in[4];
    const float* ent   = (const float*)d_in[5];
    const float* rel   = (const float*)d_in[6];
    const float* trans = (const float*)d_in[7];

    int* ws       = (int*)d_ws;
    int* counts   = ws;                 // 1024 ints
    int* offsets  = ws + 1024;          // 1024 ints
    int* cursor   = ws + 2048;          // 1024 ints
    int* bucket   = ws + 3072;          // BATCH ints
    float* scores = (float*)(ws + 3072 + BATCH);   // BATCH floats
    float* parts  = scores + BATCH;                // 64 floats
    float* out    = (float*)d_out;

    zero_counts<<<4, 256, 0, stream>>>(counts);
    count_rel<<<BATCH / 256, 256, 0, stream>>>(pos_r, counts);
    scan_rel<<<1, 1024, 0, stream>>>(counts, offsets, cursor);
    scatter_rel<<<BATCH / 256, 256, 0, stream>>>(pos_r, cursor, bucket);
    transr_rel_kernel<<<REL_TOTAL, 256, 0, stream>>>(
        pos_h, pos_t, neg_h, neg_t, ent, rel, trans,
        offsets, counts, bucket, scores);
    partial_sum<<<64, 256, 0, stream>>>(scores, parts);
    final_sum<<<1, 64, 0, stream>>>(parts, out);
}